// SAGE_36344013259391
// MI455X (gfx1250) — compile-verified
//
#include <hip/hip_runtime.h>
#include <hip/hip_bf16.h>

// ---------------------------------------------------------------------------
// GraphSAGE 2-layer forward for MI455X (gfx1250, wave32).
//   layer: mean-aggr(neighbors) @ Wl.T + bl + x @ Wr.T
// Linear commutes with mean -> transform first, aggregate in the smaller
// output dimension (halves edge-scatter traffic). Working set (~100MB) fits
// in the 192MB L2, so gathers/atomics run at L2 speed, not HBM.
// GEMMs: V_WMMA_F32_16X16X4_F32 (exact fp32, matches the fp32 reference).
// ---------------------------------------------------------------------------

typedef __attribute__((ext_vector_type(2))) float v2f;
typedef __attribute__((ext_vector_type(8))) float v8f;

// C[M x Ncols] = A[M x K] @ W^T, W[Ncols x K] row-major.
// One wave32 computes an (MREP*16)-row x (NTILES*16)-col block:
//  - B fragments loaded once per k-step, reused across MREP row strips
//  - all fragments of a k-step loaded into distinct values BEFORE the WMMA
//    burst so loads overlap and a single wait covers the group
// Out-of-range rows/cols are CLAMPED on load (WMMA rows/cols are independent;
// clamped duplicates land only in slots the guarded store discards) -> no
// divergent loads, EXEC stays all-ones around every WMMA.
template <int K, int NTILES, int MREP>
__global__ void sage_gemm_wmma_f32(const float* __restrict__ A,
                                   const float* __restrict__ W,
                                   float* __restrict__ C,
                                   int M, int Ncols) {
    const int wave = (blockIdx.x * blockDim.x + threadIdx.x) >> 5;
    const int lane = threadIdx.x & 31;
    const int strips = (M + 16 * MREP - 1) / (16 * MREP);
    if (wave >= strips) return;               // wave-uniform: EXEC all-ones below

    const int m0 = wave * (16 * MREP);
    const int half = lane >> 4;               // 0: lanes 0-15, 1: lanes 16-31
    const int lrow = lane & 15;

    const float* __restrict__ Arow[MREP];
#pragma unroll
    for (int i = 0; i < MREP; ++i) {
        int m = m0 + (i << 4) + lrow;
        if (m > M - 1) m = M - 1;             // clamp (tail block only)
        Arow[i] = A + (size_t)m * K;
    }

    const float* __restrict__ Wrow[NTILES];
#pragma unroll
    for (int t = 0; t < NTILES; ++t) {
        int n = (t << 4) + lrow;
        if (n > Ncols - 1) n = Ncols - 1;     // clamp padded columns
        Wrow[t] = W + (size_t)n * K;
    }

    v8f acc[MREP][NTILES];
#pragma unroll
    for (int i = 0; i < MREP; ++i)
#pragma unroll
        for (int t = 0; t < NTILES; ++t)
            acc[i][t] = v8f{0.f, 0.f, 0.f, 0.f, 0.f, 0.f, 0.f, 0.f};

    // A 16x4 frag: lanes 0-15 hold K=k0,k0+1; lanes 16-31 hold K=k0+2,k0+3.
    // B 4x16 frag mirrored. Contiguous pair -> one global_load_b64 each.
#pragma unroll
    for (int k0 = 0; k0 < K; k0 += 4) {
        const int kk = k0 + (half << 1);
        v2f a[MREP];
#pragma unroll
        for (int i = 0; i < MREP; ++i) a[i] = *(const v2f*)(Arow[i] + kk);
        v2f b[NTILES];
#pragma unroll
        for (int t = 0; t < NTILES; ++t) b[t] = *(const v2f*)(Wrow[t] + kk);
#pragma unroll
        for (int i = 0; i < MREP; ++i)
#pragma unroll
            for (int t = 0; t < NTILES; ++t)
                // (neg_a, A, neg_b, B, c_mod, C, reuse_a, reuse_b)
                acc[i][t] = __builtin_amdgcn_wmma_f32_16x16x4_f32(
                    false, a[i], false, b[t], (short)0, acc[i][t], false, false);
    }

    // C/D layout: VGPR r = row (strip_base + half*8 + r), column = n0 + lrow.
#pragma unroll
    for (int i = 0; i < MREP; ++i) {
#pragma unroll
        for (int t = 0; t < NTILES; ++t) {
            const int n = (t << 4) + lrow;
            if (n < Ncols) {
#pragma unroll
                for (int r = 0; r < 8; ++r) {
                    const int mr = m0 + (i << 4) + (half << 3) + r;
                    if (mr < M) C[(size_t)mr * Ncols + n] = acc[i][t][r];
                }
            }
        }
    }
}

// One wave per edge: agg[dst] += T[src] (F floats, F <= 64), optional degree count.
__global__ void sage_scatter_add(const float* __restrict__ T,
                                 const int* __restrict__ src,
                                 const int* __restrict__ dst,
                                 float* __restrict__ agg,
                                 float* __restrict__ cnt,
                                 int E, int F) {
    const int e = (blockIdx.x * blockDim.x + threadIdx.x) >> 5;
    const int lane = threadIdx.x & 31;
    if (e >= E) return;
    const int s = src[e];
    const int d = dst[e];
    const float* __restrict__ trow = T + (size_t)s * F;
    float* __restrict__ arow = agg + (size_t)d * F;
    if (lane < F)      atomicAdd(&arow[lane],      trow[lane]);   // global_atomic_add_f32
    if (lane + 32 < F) atomicAdd(&arow[lane + 32], trow[lane + 32]);
    if (cnt != nullptr && lane == 0) atomicAdd(&cnt[d], 1.0f);
}

// h = relu(agg / max(cnt,1) + bias + r), elementwise over N*F.
__global__ void sage_combine_relu(const float* __restrict__ agg,
                                  const float* __restrict__ cnt,
                                  const float* __restrict__ r,
                                  const float* __restrict__ bias,
                                  float* __restrict__ h,
                                  int N, int F) {
    const int i = blockIdx.x * blockDim.x + threadIdx.x;
    if (i >= N * F) return;
    const int node = i / F;
    const int f = i - node * F;
    float c = cnt[node];
    c = c < 1.f ? 1.f : c;
    const float v = agg[i] / c + bias[f] + r[i];
    h[i] = v > 0.f ? v : 0.f;
}

// One wave per node: o = agg/max(cnt,1) + bias + r over 40 classes, then softmax.
__global__ void sage_combine_softmax40(const float* __restrict__ agg,
                                       const float* __restrict__ cnt,
                                       const float* __restrict__ r,
                                       const float* __restrict__ bias,
                                       float* __restrict__ out,
                                       int N) {
    const int node = (blockIdx.x * blockDim.x + threadIdx.x) >> 5;
    const int lane = threadIdx.x & 31;
    if (node >= N) return;
    float c = cnt[node];
    c = c < 1.f ? 1.f : c;
    const size_t base = (size_t)node * 40;

    const float v0 = agg[base + lane] / c + bias[lane] + r[base + lane];
    const bool has2 = (lane + 32) < 40;       // lanes 0-7 carry a second class
    const float v1 = has2
        ? (agg[base + lane + 32] / c + bias[lane + 32] + r[base + lane + 32])
        : -__builtin_inff();

    float m = fmaxf(v0, v1);
#pragma unroll
    for (int o = 16; o > 0; o >>= 1) m = fmaxf(m, __shfl_xor(m, o, 32));
    const float e0 = __expf(v0 - m);
    const float e1 = has2 ? __expf(v1 - m) : 0.f;
    float s = e0 + e1;
#pragma unroll
    for (int o = 16; o > 0; o >>= 1) s += __shfl_xor(s, o, 32);
    const float inv = 1.0f / s;

    out[base + lane] = e0 * inv;
    if (has2) out[base + lane + 32] = e1 * inv;
}

extern "C" void kernel_launch(void* const* d_in, const int* in_sizes, int n_in,
                              void* d_out, int out_size, void* d_ws, size_t ws_size,
                              hipStream_t stream) {
    (void)n_in; (void)out_size; (void)ws_size;

    const int IN = 128, H = 64, OUT = 40;
    const int N = in_sizes[0] / IN;           // 100000
    const int E = in_sizes[1] / 2;            // 1600000

    const float* x   = (const float*)d_in[0];
    const int*   ei  = (const int*)d_in[1];
    const float* Wl1 = (const float*)d_in[2];
    const float* bl1 = (const float*)d_in[3];
    const float* Wr1 = (const float*)d_in[4];
    const float* Wl2 = (const float*)d_in[5];
    const float* bl2 = (const float*)d_in[6];
    const float* Wr2 = (const float*)d_in[7];
    float* out = (float*)d_out;

    const int* src = ei;        // edge_index[0]
    const int* dst = ei + E;    // edge_index[1]

    // Workspace: 4 x (N*H floats) + cnt (N floats)  ~= 103 MB
    float* t1  = (float*)d_ws;                // layer1: x @ Wl1.T          [N,64]
    float* r1  = t1 + (size_t)N * H;          // layer1: x @ Wr1.T          [N,64]
    float* agg = r1 + (size_t)N * H;          // aggregation buffer (reused)[N,64]
    float* h   = agg + (size_t)N * H;         // hidden activations         [N,64]
    float* cnt = h + (size_t)N * H;           // in-degree counts           [N]
    // layer2 reuses: t2 -> t1 slot [N,40], r2 -> r1 slot [N,40], agg slot [N,40]

    const int THREADS = 256;                  // 8 wave32s per block
    const int WPB = THREADS / 32;
    const int MREP = 2;                       // 16-row strips per wave
    const int strips = (N + 16 * MREP - 1) / (16 * MREP);   // 3125
    const dim3 gemm_grid((strips + WPB - 1) / WPB);

    // ---- layer 1 ----
    hipMemsetAsync(agg, 0, (size_t)N * H * sizeof(float), stream);
    hipMemsetAsync(cnt, 0, (size_t)N * sizeof(float), stream);

    sage_gemm_wmma_f32<128, 4, 2><<<gemm_grid, THREADS, 0, stream>>>(x, Wl1, t1, N, H);
    sage_gemm_wmma_f32<128, 4, 2><<<gemm_grid, THREADS, 0, stream>>>(x, Wr1, r1, N, H);

    sage_scatter_add<<<(E + WPB - 1) / WPB, THREADS, 0, stream>>>(t1, src, dst, agg, cnt, E, H);
    sage_combine_relu<<<(N * H + THREADS - 1) / THREADS, THREADS, 0, stream>>>(
        agg, cnt, r1, bl1, h, N, H);

    // ---- layer 2 ----
    hipMemsetAsync(agg, 0, (size_t)N * OUT * sizeof(float), stream);

    sage_gemm_wmma_f32<64, 3, 2><<<gemm_grid, THREADS, 0, stream>>>(h, Wl2, t1, N, OUT);
    sage_gemm_wmma_f32<64, 3, 2><<<gemm_grid, THREADS, 0, stream>>>(h, Wr2, r1, N, OUT);

    sage_scatter_add<<<(E + WPB - 1) / WPB, THREADS, 0, stream>>>(t1, src, dst, agg, nullptr, E, OUT);
    sage_combine_softmax40<<<(N + WPB - 1) / WPB, THREADS, 0, stream>>>(
        agg, cnt, r1, bl2, out, N);
}